// GAT_49263274885562
// MI455X (gfx1250) — compile-verified
//
#include <hip/hip_runtime.h>
#include <hip/hip_bf16.h>

typedef float v2f __attribute__((ext_vector_type(2)));
typedef float v8f __attribute__((ext_vector_type(8)));

#define NEG_SLOPE 0.2f
#define EPSX 1e-16f

// ---- order-preserving float<->uint encoding for atomicMax-based segment max ----
__device__ __forceinline__ unsigned encf(float f) {
  unsigned u = __float_as_uint(f);
  return (u & 0x80000000u) ? ~u : (u | 0x80000000u);
}
__device__ __forceinline__ float decf(unsigned e) {
  unsigned v = (e & 0x80000000u) ? (e & 0x7FFFFFFFu) : ~e;
  return __uint_as_float(v);
}
#define ENC_NEG_INF 0x007FFFFFu  // encf(-inf)

__device__ __forceinline__ float lrelu(float x) { return x > 0.f ? x : NEG_SLOPE * x; }

__device__ __forceinline__ float wave_sum(float v) {
#pragma unroll
  for (int m = 16; m >= 1; m >>= 1) v += __shfl_xor(v, m, 32);
  return v;
}

__device__ __forceinline__ void edge_sd(const long long* __restrict__ ei, int E, int e,
                                        int& s, int& d) {
  if (e < E) {
    s = (int)ei[e];
    d = (int)ei[(size_t)E + e];
  } else {  // self loops appended
    s = d = e - E;
  }
}

// ---------------- init: zero accumulators, set segment-max to -inf ----------------
__global__ void k_init(float* hagg1, float* denom1, unsigned* amax1, float* denom2,
                       unsigned* amax2, float* out, int N) {
  size_t total = (size_t)N * 256;
  size_t stride = (size_t)gridDim.x * blockDim.x;
  for (size_t i = (size_t)blockIdx.x * blockDim.x + threadIdx.x; i < total; i += stride) {
    hagg1[i] = 0.f;
    if (i < (size_t)N * 4) { denom1[i] = 0.f; amax1[i] = ENC_NEG_INF; }
    if (i < (size_t)N) { denom2[i] = 0.f; amax2[i] = ENC_NEG_INF; }
    if (i < (size_t)N * 2) out[i] = 0.f;
  }
}

// ---------------- layer-1 GEMM: h1[N,256] = x[N,128] @ W1[256,128]^T ----------------
// One wave computes one 16x16 tile with V_WMMA_F32_16X16X4_F32, K-loop of 32.
__global__ void k_gemm1(const float* __restrict__ x, const float* __restrict__ W1,
                        float* __restrict__ h1, int N) {
  int wave = threadIdx.x >> 5;
  int lane = threadIdx.x & 31;
  int mBase = blockIdx.x * 16;
  int nBase = (blockIdx.y * 4 + wave) * 16;
  int hf = lane >> 4;   // lane half selects K pair {0,1} vs {2,3}
  int lo = lane & 15;   // row (A) / col (B) within tile
  int arow = mBase + lo;
  if (arow >= N) arow = N - 1;  // safe clamp (N=50000 is divisible by 16 anyway)
  const float* xr = x + (size_t)arow * 128 + 2 * hf;
  const float* wr = W1 + (size_t)(nBase + lo) * 128 + 2 * hf;
  v8f c = {};
#pragma unroll
  for (int kk = 0; kk < 128; kk += 4) {
    v2f a = *(const v2f*)(xr + kk);  // A: M=lo, K=kk+2*hf+{0,1}
    v2f b = *(const v2f*)(wr + kk);  // B: N=lo, K=kk+2*hf+{0,1}  (B[k][n] = W1[n][k])
    c = __builtin_amdgcn_wmma_f32_16x16x4_f32(false, a, false, b, (short)0, c, false, false);
  }
  // C/D layout: VGPR r -> row mBase + r + 8*hf, col nBase + lo
#pragma unroll
  for (int r = 0; r < 8; ++r) {
    int orow = mBase + r + 8 * hf;
    if (orow < N) h1[(size_t)orow * 256 + nBase + lo] = c[r];
  }
}

// ---------------- per-node attention logits: a_src/a_dst [N,4] ----------------
__global__ void k_att1(const float* __restrict__ h1, const float* __restrict__ att_src,
                       const float* __restrict__ att_dst, float* a_src, float* a_dst, int N) {
  int i = (int)((blockIdx.x * (size_t)blockDim.x + threadIdx.x) >> 5);
  int lane = threadIdx.x & 31;
  if (i >= N) return;
  const float* hr = h1 + (size_t)i * 256;
#pragma unroll
  for (int h = 0; h < 4; ++h) {
    float v0 = hr[h * 64 + lane], v1 = hr[h * 64 + 32 + lane];
    float s = v0 * att_src[h * 64 + lane] + v1 * att_src[h * 64 + 32 + lane];
    float d = v0 * att_dst[h * 64 + lane] + v1 * att_dst[h * 64 + 32 + lane];
    s = wave_sum(s);
    d = wave_sum(d);
    if (lane == 0) { a_src[(size_t)i * 4 + h] = s; a_dst[(size_t)i * 4 + h] = d; }
  }
}

// ---------------- edge pass 1a: segment max of leaky-relu logits ----------------
__global__ void k_edge_max1(const long long* __restrict__ ei, int E, int ET,
                            const float* __restrict__ a_src, const float* __restrict__ a_dst,
                            unsigned* amax) {
  int e = blockIdx.x * blockDim.x + threadIdx.x;
  if (e >= ET) return;
  int s, d;
  edge_sd(ei, E, e, s, d);
  float4 as = *(const float4*)(a_src + (size_t)s * 4);
  float4 ad = *(const float4*)(a_dst + (size_t)d * 4);
  unsigned* am = amax + (size_t)d * 4;
  atomicMax(am + 0, encf(lrelu(as.x + ad.x)));
  atomicMax(am + 1, encf(lrelu(as.y + ad.y)));
  atomicMax(am + 2, encf(lrelu(as.z + ad.z)));
  atomicMax(am + 3, encf(lrelu(as.w + ad.w)));
}

// ---------------- edge pass 1b: segment sum of exp(alpha - amax) ----------------
__global__ void k_edge_sum1(const long long* __restrict__ ei, int E, int ET,
                            const float* __restrict__ a_src, const float* __restrict__ a_dst,
                            const unsigned* __restrict__ amax, float* denom) {
  int e = blockIdx.x * blockDim.x + threadIdx.x;
  if (e >= ET) return;
  int s, d;
  edge_sd(ei, E, e, s, d);
  float4 as = *(const float4*)(a_src + (size_t)s * 4);
  float4 ad = *(const float4*)(a_dst + (size_t)d * 4);
  const unsigned* am = amax + (size_t)d * 4;
  float* dn = denom + (size_t)d * 4;
  atomicAdd(dn + 0, __expf(lrelu(as.x + ad.x) - decf(am[0])));
  atomicAdd(dn + 1, __expf(lrelu(as.y + ad.y) - decf(am[1])));
  atomicAdd(dn + 2, __expf(lrelu(as.z + ad.z) - decf(am[2])));
  atomicAdd(dn + 3, __expf(lrelu(as.w + ad.w) - decf(am[3])));
}

// ---------------- edge pass 1c: wave-per-edge weighted scatter-add ----------------
// Lane L handles channels [L*8, L*8+8) -> head = L>>3; float4 gathers + f32 atomics.
__global__ void k_edge_agg1(const long long* __restrict__ ei, int E, int ET,
                            const float* __restrict__ a_src, const float* __restrict__ a_dst,
                            const unsigned* __restrict__ amax, const float* __restrict__ denom,
                            const float* __restrict__ h1, float* hagg) {
  int e = (int)((blockIdx.x * (size_t)blockDim.x + threadIdx.x) >> 5);
  int lane = threadIdx.x & 31;
  if (e >= ET) return;
  int s, d;
  edge_sd(ei, E, e, s, d);
  int h = lane >> 3;
  float alpha = lrelu(a_src[(size_t)s * 4 + h] + a_dst[(size_t)d * 4 + h]);
  float ex = __expf(alpha - decf(amax[(size_t)d * 4 + h]));
  float coef = ex / (denom[(size_t)d * 4 + h] + EPSX);
  const float4* hp = (const float4*)(h1 + (size_t)s * 256 + lane * 8);
  float4 v0 = hp[0], v1 = hp[1];
  float* op = hagg + (size_t)d * 256 + lane * 8;
  atomicAdd(op + 0, v0.x * coef);
  atomicAdd(op + 1, v0.y * coef);
  atomicAdd(op + 2, v0.z * coef);
  atomicAdd(op + 3, v0.w * coef);
  atomicAdd(op + 4, v1.x * coef);
  atomicAdd(op + 5, v1.y * coef);
  atomicAdd(op + 6, v1.z * coef);
  atomicAdd(op + 7, v1.w * coef);
}

// ---------------- node pass: +b1, ReLU, tiny GEMM2 (256->2), layer-2 logits ----------------
__global__ void k_node2(const float* __restrict__ hagg, const float* __restrict__ b1,
                        const float* __restrict__ W2, const float* __restrict__ as2,
                        const float* __restrict__ ad2, float* h2, float* a_src2, float* a_dst2,
                        int N) {
  int i = (int)((blockIdx.x * (size_t)blockDim.x + threadIdx.x) >> 5);
  int lane = threadIdx.x & 31;
  if (i >= N) return;
  const float* hr = hagg + (size_t)i * 256 + lane * 8;
  const float* br = b1 + lane * 8;
  const float* w0 = W2 + lane * 8;
  const float* w1 = W2 + 256 + lane * 8;
  float d0 = 0.f, d1 = 0.f;
#pragma unroll
  for (int j = 0; j < 8; ++j) {
    float v = hr[j] + br[j];
    v = v > 0.f ? v : 0.f;  // relu between layers
    d0 += v * w0[j];
    d1 += v * w1[j];
  }
  d0 = wave_sum(d0);
  d1 = wave_sum(d1);
  if (lane == 0) {
    h2[(size_t)i * 2 + 0] = d0;
    h2[(size_t)i * 2 + 1] = d1;
    a_src2[i] = d0 * as2[0] + d1 * as2[1];
    a_dst2[i] = d0 * ad2[0] + d1 * ad2[1];
  }
}

// ---------------- layer-2 edge passes (H=1, C=2) ----------------
__global__ void k_edge_max2(const long long* __restrict__ ei, int E, int ET,
                            const float* __restrict__ a_src, const float* __restrict__ a_dst,
                            unsigned* amax) {
  int e = blockIdx.x * blockDim.x + threadIdx.x;
  if (e >= ET) return;
  int s, d;
  edge_sd(ei, E, e, s, d);
  atomicMax(amax + d, encf(lrelu(a_src[s] + a_dst[d])));
}

__global__ void k_edge_sum2(const long long* __restrict__ ei, int E, int ET,
                            const float* __restrict__ a_src, const float* __restrict__ a_dst,
                            const unsigned* __restrict__ amax, float* denom) {
  int e = blockIdx.x * blockDim.x + threadIdx.x;
  if (e >= ET) return;
  int s, d;
  edge_sd(ei, E, e, s, d);
  atomicAdd(denom + d, __expf(lrelu(a_src[s] + a_dst[d]) - decf(amax[d])));
}

__global__ void k_edge_agg2(const long long* __restrict__ ei, int E, int ET,
                            const float* __restrict__ a_src, const float* __restrict__ a_dst,
                            const unsigned* __restrict__ amax, const float* __restrict__ denom,
                            const float* __restrict__ h2, float* out) {
  int e = blockIdx.x * blockDim.x + threadIdx.x;
  if (e >= ET) return;
  int s, d;
  edge_sd(ei, E, e, s, d);
  float ex = __expf(lrelu(a_src[s] + a_dst[d]) - decf(amax[d]));
  float coef = ex / (denom[d] + EPSX);
  atomicAdd(out + (size_t)d * 2 + 0, h2[(size_t)s * 2 + 0] * coef);
  atomicAdd(out + (size_t)d * 2 + 1, h2[(size_t)s * 2 + 1] * coef);
}

__global__ void k_bias2(float* out, const float* __restrict__ b2, int N2) {
  int i = blockIdx.x * blockDim.x + threadIdx.x;
  if (i < N2) out[i] += b2[i & 1];
}

extern "C" void kernel_launch(void* const* d_in, const int* in_sizes, int n_in, void* d_out,
                              int out_size, void* d_ws, size_t ws_size, hipStream_t stream) {
  const float* x = (const float*)d_in[0];
  const long long* ei = (const long long*)d_in[1];  // int64 edge_index [2,E]
  const float* W1 = (const float*)d_in[2];
  const float* as1 = (const float*)d_in[3];
  const float* ad1 = (const float*)d_in[4];
  const float* b1 = (const float*)d_in[5];
  const float* W2 = (const float*)d_in[6];
  const float* as2 = (const float*)d_in[7];
  const float* ad2 = (const float*)d_in[8];
  const float* b2 = (const float*)d_in[9];

  int N = in_sizes[0] / 128;  // 50000
  int E = in_sizes[1] / 2;    // 800000
  int ET = E + N;             // + self loops

  // workspace layout (all 16B aligned for N=50000)
  float* ws = (float*)d_ws;
  size_t o = 0;
  float* h1 = ws + o;      o += (size_t)N * 256;
  float* hagg1 = ws + o;   o += (size_t)N * 256;
  float* a_src1 = ws + o;  o += (size_t)N * 4;
  float* a_dst1 = ws + o;  o += (size_t)N * 4;
  unsigned* amax1 = (unsigned*)(ws + o); o += (size_t)N * 4;
  float* denom1 = ws + o;  o += (size_t)N * 4;
  float* h2 = ws + o;      o += (size_t)N * 2;
  float* a_src2 = ws + o;  o += (size_t)N;
  float* a_dst2 = ws + o;  o += (size_t)N;
  unsigned* amax2 = (unsigned*)(ws + o); o += (size_t)N;
  float* denom2 = ws + o;  o += (size_t)N;
  float* out = (float*)d_out;

  k_init<<<2048, 256, 0, stream>>>(hagg1, denom1, amax1, denom2, amax2, out, N);

  dim3 gG((N + 15) / 16, 4);
  k_gemm1<<<gG, 128, 0, stream>>>(x, W1, h1, N);

  int nodeBlocks = (N + 7) / 8;  // 8 waves/block, wave per node
  k_att1<<<nodeBlocks, 256, 0, stream>>>(h1, as1, ad1, a_src1, a_dst1, N);

  int edgeBlocks = (ET + 255) / 256;  // thread per edge
  k_edge_max1<<<edgeBlocks, 256, 0, stream>>>(ei, E, ET, a_src1, a_dst1, amax1);
  k_edge_sum1<<<edgeBlocks, 256, 0, stream>>>(ei, E, ET, a_src1, a_dst1, amax1, denom1);

  int aggBlocks = (ET + 7) / 8;  // wave per edge
  k_edge_agg1<<<aggBlocks, 256, 0, stream>>>(ei, E, ET, a_src1, a_dst1, amax1, denom1, h1,
                                             hagg1);

  k_node2<<<nodeBlocks, 256, 0, stream>>>(hagg1, b1, W2, as2, ad2, h2, a_src2, a_dst2, N);

  k_edge_max2<<<edgeBlocks, 256, 0, stream>>>(ei, E, ET, a_src2, a_dst2, amax2);
  k_edge_sum2<<<edgeBlocks, 256, 0, stream>>>(ei, E, ET, a_src2, a_dst2, amax2, denom2);
  k_edge_agg2<<<edgeBlocks, 256, 0, stream>>>(ei, E, ET, a_src2, a_dst2, amax2, denom2, h2,
                                              out);

  k_bias2<<<(2 * N + 255) / 256, 256, 0, stream>>>(out, b2, 2 * N);
}